// LMNNLoss_SP_7146825581134
// MI455X (gfx1250) — compile-verified
//
#include <hip/hip_runtime.h>

typedef __attribute__((ext_vector_type(2))) float v2f;
typedef __attribute__((ext_vector_type(8))) float v8f;

#define NSEG 2048
#define PPS  512
#define DIM  128
#define NPTS 511   // points per segment excluding the center

// One workgroup (256 thr = 8 waves) per segment.
// Each wave computes d2 for 4 tiles of 16 points via WMMA Gram-diagonal.
__launch_bounds__(256)
__global__ void lmnn_segment_kernel(const float* __restrict__ outputs,
                                    const int* __restrict__ labels,
                                    float* __restrict__ seg_loss) {
  const int s    = blockIdx.x;
  const int t    = threadIdx.x;
  const int lane = t & 31;
  const int wave = t >> 5;

  __shared__ __align__(16) float lds_center[DIM];
  __shared__ float lds_d2[PPS];
  __shared__ int   lds_same[PPS];
  __shared__ float lds_red[256];

  const float* seg_base = outputs + (size_t)s * PPS * DIM;
  const int*   lab_base = labels  + (size_t)s * NPTS;

  // Stage center row (outputs[s,0,:]) into LDS.
  if (t < DIM) lds_center[t] = seg_base[t];

  // same[j] = (label[j] == label[0]); pad slot 511 -> not-same.
  const int lab0 = lab_base[0];
  for (int jj = t; jj < PPS; jj += 256) {
    lds_same[jj] = (jj < NPTS) ? (lab_base[jj] == lab0) : 0;
  }
  __syncthreads();

  const int  row_m = lane & 15;
  const bool hi    = lane >= 16;
  const int  koff  = hi ? 4 : 0;

  for (int tile = wave * 4; tile < wave * 4 + 4; ++tile) {
    const int j       = tile * 16 + row_m;           // pts index 0..511 (511 = pad)
    const int src_row = (j < NPTS) ? (j + 1) : 0;    // clamp pad row (data unused)
    const float* rowp = seg_base + (size_t)src_row * DIM;

    v8f acc = {};
    #pragma unroll
    for (int ks = 0; ks < 16; ++ks) {
      const int off = ks * 8 + koff;                 // 16B-aligned K-slice
      __builtin_prefetch(rowp + off + 8, 0, 0);      // global_prefetch_b8 (next chunk)
      const float4 p = *(const float4*)(rowp + off);
      const float4 c = *(const float4*)(lds_center + off);
      v2f x0; x0.x = p.x - c.x; x0.y = p.y - c.y;
      v2f x1; x1.x = p.z - c.z; x1.y = p.w - c.w;
      // D += X * X^T : A(16x4) and B(4x16) layouts are mutual transposes,
      // so the same per-lane registers serve as both operands.
      acc = __builtin_amdgcn_wmma_f32_16x16x4_f32(false, x0, false, x0,
                                                  (short)0, acc, false, false);
      acc = __builtin_amdgcn_wmma_f32_16x16x4_f32(false, x1, false, x1,
                                                  (short)0, acc, false, false);
    }

    // Diagonal of 16x16 f32 D: (m,m) at {VGPR m, lane m} for m<8,
    // {VGPR m-8, lane m+16} for m>=8.
    float dval = 0.f;
    #pragma unroll
    for (int r = 0; r < 8; ++r) {
      if (lane == r)      dval = acc[r];
      if (lane == 24 + r) dval = acc[r];
    }
    if (lane < 8 || lane >= 24) {
      const int m  = (lane < 8) ? lane : (lane - 16);
      const int jj = tile * 16 + m;
      lds_d2[jj] = (jj < NPTS) ? dval : __builtin_inff();
    }
  }
  __syncthreads();

  // pull = min over same-labeled of d2 (j=0 is always same -> min exists).
  float v0 = lds_same[t]       ? lds_d2[t]       : __builtin_inff();
  float v1 = lds_same[t + 256] ? lds_d2[t + 256] : __builtin_inff();
  lds_red[t] = fminf(v0, v1);
  __syncthreads();
  for (int stp = 128; stp >= 1; stp >>= 1) {
    if (t < stp) lds_red[t] = fminf(lds_red[t], lds_red[t + stp]);
    __syncthreads();
  }
  const float pull   = lds_red[0];
  const float margin = 1.0f + pull;
  __syncthreads();

  // push = sum over differently-labeled of max(margin - d2, 0).
  // Pad slot: same=0, d2=+inf -> max(-inf,0)=0.
  float push = 0.f;
  if (!lds_same[t])       push += fmaxf(margin - lds_d2[t],       0.f);
  if (!lds_same[t + 256]) push += fmaxf(margin - lds_d2[t + 256], 0.f);
  lds_red[t] = push;
  __syncthreads();
  for (int stp = 128; stp >= 1; stp >>= 1) {
    if (t < stp) lds_red[t] += lds_red[t + stp];
    __syncthreads();
  }
  if (t == 0) seg_loss[s] = pull + lds_red[0];
}

// Deterministic fixed-order final reduction of 2048 per-segment losses.
__launch_bounds__(256)
__global__ void lmnn_final_reduce(const float* __restrict__ seg_loss,
                                  float* __restrict__ out) {
  __shared__ float red[256];
  const int t = threadIdx.x;
  float acc = 0.f;
  #pragma unroll
  for (int i = 0; i < NSEG / 256; ++i) acc += seg_loss[t + i * 256];
  red[t] = acc;
  __syncthreads();
  for (int stp = 128; stp >= 1; stp >>= 1) {
    if (t < stp) red[t] += red[t + stp];
    __syncthreads();
  }
  if (t == 0) out[0] = red[0] * (1.0f / ((float)NSEG * (float)PPS));
}

extern "C" void kernel_launch(void* const* d_in, const int* in_sizes, int n_in,
                              void* d_out, int out_size, void* d_ws, size_t ws_size,
                              hipStream_t stream) {
  const float* outputs = (const float*)d_in[0];
  const int*   labels  = (const int*)d_in[1];
  float* seg_loss = (float*)d_ws;              // 2048 floats of scratch

  lmnn_segment_kernel<<<NSEG, 256, 0, stream>>>(outputs, labels, seg_loss);
  lmnn_final_reduce<<<1, 256, 0, stream>>>(seg_loss, (float*)d_out);
}